// GCN_67645734912112
// MI455X (gfx1250) — compile-verified
//
#include <hip/hip_runtime.h>
#include <hip/hip_bf16.h>

// ---------------------------------------------------------------------------
// GCN on MI455X (gfx1250, wave32).
// GEMMs use V_WMMA_F32_16X16X4_F32 (full fp32 on the matrix pipe); the
// edge scatter/gather phase dominates (~17GB traffic, L2-resident since one
// 100k x 256 f32 feature buffer = 102.4MB < 192MB L2).
// Bias before BatchNorm cancels analytically, so only the last-layer bias
// (pre log-softmax) is applied.
// ---------------------------------------------------------------------------

typedef __attribute__((ext_vector_type(2))) float v2f;
typedef __attribute__((ext_vector_type(8))) float v8f;

#define DIN   128
#define DH    256
#define DOUT  40
#define BN_EPS 1e-5f

static __device__ __forceinline__ void atomic_add_f32(float* p, float v) {
    __hip_atomic_fetch_add(p, v, __ATOMIC_RELAXED, __HIP_MEMORY_SCOPE_AGENT);
}

// -------------------------------- utility ----------------------------------
__global__ void k_fill(float* __restrict__ p, float v, size_t n) {
    size_t i = (size_t)blockIdx.x * blockDim.x + threadIdx.x;
    if (i < n) p[i] = v;
}

__global__ void k_deg(const long long* __restrict__ ei, float* __restrict__ deg,
                      long long nE) {
    size_t e = (size_t)blockIdx.x * blockDim.x + threadIdx.x;
    if (e >= (size_t)nE) return;
    atomic_add_f32(&deg[(int)ei[e]], 1.0f);
}

__global__ void k_rsqrt_inplace(float* __restrict__ p, int n) {
    int i = blockIdx.x * blockDim.x + threadIdx.x;
    if (i < n) p[i] = rsqrtf(p[i]);
}

// ------------------------------ WMMA GEMM ----------------------------------
// H[nRows x Nout] = X[nRows x K] * W[K x Nout], fp32 via V_WMMA_F32_16X16X4_F32.
// Block: 256 threads = 8 waves; 16-row A tile staged in LDS (pitch K+4 ->
// bank-conflict-free b64 frag loads); wave w owns output cols [nbase,nbase+16).
// Out-of-range output columns are CLAMPED (not zeroed): WMMA derives output
// column j only from B column j, so a clamped lane only corrupts columns that
// the guarded store never writes. This keeps the inner loop branch-free.
__global__ __launch_bounds__(256)
void k_gemm_wmma(const float* __restrict__ X, const float* __restrict__ W,
                 float* __restrict__ H, int nRows, int K, int Nout)
{
    __shared__ float As[16 * (256 + 4)];          // max K = 256
    const int P   = K + 4;                        // LDS row pitch (floats)
    const int tid = threadIdx.x;
    const int mbase = blockIdx.x * 16;

    // cooperative float4 stage of the 16 x K A tile
    const int K4 = K >> 2;
    for (int i = tid; i < 16 * K4; i += 256) {
        int r = i / K4, kc = i - r * K4;
        int g = mbase + r;
        float4 v = make_float4(0.f, 0.f, 0.f, 0.f);
        if (g < nRows) v = *reinterpret_cast<const float4*>(&X[(size_t)g * K + kc * 4]);
        *reinterpret_cast<float4*>(&As[r * P + kc * 4]) = v;
    }
    __syncthreads();

    const int wave = tid >> 5;
    const int lane = tid & 31;
    const int half = lane >> 4;                   // 0: K{0,1}, 1: K{2,3}
    const int l16  = lane & 15;
    const int nbase = (blockIdx.y * 8 + wave) * 16;
    if (nbase >= Nout) return;                    // wave fully out of range

    const int  n   = nbase + l16;
    const bool nok = (n < Nout);
    const int  nc  = nok ? n : (Nout - 1);        // clamped column (see above)

    // pointer-walked B addressing: no 64-bit multiplies in the loop
    const float* __restrict__ wp = W + nc + (size_t)(2 * half) * Nout;
    const size_t stepW = (size_t)4 * Nout;

    v8f acc = {0.f, 0.f, 0.f, 0.f, 0.f, 0.f, 0.f, 0.f};
    const float* __restrict__ arow = &As[l16 * P + 2 * half];

#pragma unroll 4
    for (int k0 = 0; k0 < K; k0 += 4) {
        // A frag: 16x4 f32 layout — lanes<16 hold K{0,1}, lanes>=16 K{2,3}
        v2f a = *reinterpret_cast<const v2f*>(&arow[k0]);
        // B frag: 4x16 f32 — VGPR0: K=0/K=2 across lane halves, VGPR1: K=1/K=3
        v2f b;
        b.x = wp[0];
        b.y = wp[Nout];
        acc = __builtin_amdgcn_wmma_f32_16x16x4_f32(
                  false, a, false, b, (short)0, acc, false, false);
        wp += stepW;
    }

    // C/D layout: VGPR r -> row mbase + r + 8*half, col = nbase + l16
    const int m0 = mbase + 8 * half;
#pragma unroll
    for (int r = 0; r < 8; ++r) {
        int m = m0 + r;
        if (m < nRows && nok) H[(size_t)m * Nout + n] = acc[r];
    }
}

// ------------------------- aggregation (A + I, normalized) -----------------
// AGG[i,:] = dinv[i]^2 * H[i,:]   (self-loop term; also inits poisoned ws)
template <int C>
__global__ __launch_bounds__(256)
void k_selfloop(const float* __restrict__ H, const float* __restrict__ dinv,
                float* __restrict__ AGG, int nNodes)
{
    constexpr int c4n = C >> 2;
    size_t idx = (size_t)blockIdx.x * blockDim.x + threadIdx.x;
    size_t total = (size_t)nNodes * c4n;
    if (idx >= total) return;
    int i = (int)(idx / c4n);
    float dv = dinv[i];
    float s = dv * dv;
    float4 h = reinterpret_cast<const float4*>(H)[idx];
    reinterpret_cast<float4*>(AGG)[idx] = make_float4(h.x * s, h.y * s, h.z * s, h.w * s);
}

// AGG[row,:] += dinv[row]*dinv[col] * H[col,:]  (edge-parallel, L2 atomics)
template <int C>
__global__ __launch_bounds__(256)
void k_scatter(const long long* __restrict__ ei, const float* __restrict__ H,
               const float* __restrict__ dinv, float* __restrict__ AGG,
               long long nE)
{
    constexpr int c4n = C >> 2;
    size_t t = (size_t)blockIdx.x * blockDim.x + threadIdx.x;
    size_t total = (size_t)nE * c4n;
    if (t >= total) return;
    size_t e  = t / c4n;                  // compile-time divisor -> mul/shift
    int    c4 = (int)(t - e * c4n);
    int r = (int)ei[e];
    int c = (int)ei[(size_t)nE + e];
    float nrm = dinv[r] * dinv[c];
    float4 h = *reinterpret_cast<const float4*>(&H[(size_t)c * C + c4 * 4]);
    float* dst = &AGG[(size_t)r * C + c4 * 4];
    atomic_add_f32(dst + 0, h.x * nrm);
    atomic_add_f32(dst + 1, h.y * nrm);
    atomic_add_f32(dst + 2, h.z * nrm);
    atomic_add_f32(dst + 3, h.w * nrm);
}

// ------------------------------ BatchNorm ----------------------------------
// thread c of each block owns channel c; coalesced column partial sums.
__global__ __launch_bounds__(256)
void k_bn_stats(const float* __restrict__ F, float* __restrict__ sum,
                float* __restrict__ ssum, int nNodes, int rowsPerBlock)
{
    int c  = threadIdx.x;                        // C == 256
    int r0 = blockIdx.x * rowsPerBlock;
    int r1 = min(r0 + rowsPerBlock, nNodes);
    float s = 0.f, q = 0.f;
    for (int r = r0; r < r1; ++r) {
        float v = F[(size_t)r * DH + c];
        s += v; q += v * v;
    }
    atomic_add_f32(&sum[c], s);
    atomic_add_f32(&ssum[c], q);
}

__global__ void k_bn_apply_relu(float* __restrict__ F, const float* __restrict__ sum,
                                const float* __restrict__ ssum,
                                const float* __restrict__ gamma,
                                const float* __restrict__ beta, int nNodes)
{
    size_t idx = (size_t)blockIdx.x * blockDim.x + threadIdx.x;
    size_t total = (size_t)nNodes * DH;
    if (idx >= total) return;
    int c = (int)(idx & (DH - 1));
    float invN = 1.0f / (float)nNodes;
    float mu  = sum[c] * invN;
    float var = ssum[c] * invN - mu * mu;       // biased variance (ddof=0)
    float v = (F[idx] - mu) * rsqrtf(var + BN_EPS) * gamma[c] + beta[c];
    F[idx] = fmaxf(v, 0.f);
}

// --------------------------- final log-softmax -----------------------------
// one wave32 per node; lanes cover cols {lane, lane+32}, shfl_xor reductions.
__global__ __launch_bounds__(256)
void k_bias_logsoftmax(const float* __restrict__ Z, const float* __restrict__ bl,
                       float* __restrict__ out, int nNodes)
{
    int node = blockIdx.x * 8 + (threadIdx.x >> 5);
    int lane = threadIdx.x & 31;
    if (node >= nNodes) return;
    const float* z = &Z[(size_t)node * DOUT];
    float z1 = z[lane] + bl[lane];
    float z2 = (lane < DOUT - 32) ? (z[32 + lane] + bl[32 + lane]) : -3.0e38f;
    float m = fmaxf(z1, z2);
    for (int o = 16; o; o >>= 1) m = fmaxf(m, __shfl_xor(m, o, 32));
    float s = expf(z1 - m) + ((lane < DOUT - 32) ? expf(z2 - m) : 0.f);
    for (int o = 16; o; o >>= 1) s += __shfl_xor(s, o, 32);
    float l = m + logf(s);
    float* o_ = &out[(size_t)node * DOUT];
    o_[lane] = z1 - l;
    if (lane < DOUT - 32) o_[32 + lane] = z2 - l;
}

// ------------------------------- launcher ----------------------------------
extern "C" void kernel_launch(void* const* d_in, const int* in_sizes, int n_in,
                              void* d_out, int out_size, void* d_ws, size_t ws_size,
                              hipStream_t stream)
{
    const float*     x     = (const float*)d_in[0];
    const float*     W0    = (const float*)d_in[1];
    // d_in[2] = b0, d_in[4] = bh: cancelled by BatchNorm, unused.
    const float*     Wh    = (const float*)d_in[3];   // [3, 256, 256]
    const float*     gamma = (const float*)d_in[5];   // [4, 256]
    const float*     beta  = (const float*)d_in[6];   // [4, 256]
    const float*     Wl    = (const float*)d_in[7];   // [256, 40]
    const float*     bl    = (const float*)d_in[8];   // [40]
    const long long* ei    = (const long long*)d_in[9]; // [2, E] int64

    const int       nN = in_sizes[0] / DIN;
    const long long nE = in_sizes[9] / 2;

    float* ws   = (float*)d_ws;
    float* dinv = ws;                       // [nN]
    float* sSum = ws + nN;                  // [256]
    float* sSq  = sSum + DH;                // [256]
    size_t featOff = ((size_t)nN + 2 * DH + 3) & ~(size_t)3;   // 16B align
    float* B1 = ws + featOff;               // [nN*256] features / agg
    float* B2 = B1 + (size_t)nN * DH;       // [nN*256] messages h
    (void)n_in; (void)out_size; (void)ws_size;

    // ---- normalized degrees: deg = 1 (self-loop) + count(row); dinv = rsqrt ----
    k_fill<<<(nN + 255) / 256, 256, 0, stream>>>(dinv, 1.0f, (size_t)nN);
    k_deg<<<(unsigned)((nE + 255) / 256), 256, 0, stream>>>(ei, dinv, nE);
    k_rsqrt_inplace<<<(nN + 255) / 256, 256, 0, stream>>>(dinv, nN);

    auto run_conv = [&](const float* Xin, int K, const float* Wt, int Cout,
                        float* Hmsg, float* Aout) {
        dim3 g((nN + 15) / 16, (Cout + 127) / 128);
        k_gemm_wmma<<<g, 256, 0, stream>>>(Xin, Wt, Hmsg, nN, K, Cout);
        int c4 = Cout >> 2;
        size_t tn = (size_t)nN * c4;
        size_t te = (size_t)nE * c4;
        unsigned gn = (unsigned)((tn + 255) / 256);
        unsigned ge = (unsigned)((te + 255) / 256);
        if (Cout == DH) {
            k_selfloop<DH><<<gn, 256, 0, stream>>>(Hmsg, dinv, Aout, nN);
            k_scatter<DH><<<ge, 256, 0, stream>>>(ei, Hmsg, dinv, Aout, nE);
        } else {
            k_selfloop<DOUT><<<gn, 256, 0, stream>>>(Hmsg, dinv, Aout, nN);
            k_scatter<DOUT><<<ge, 256, 0, stream>>>(ei, Hmsg, dinv, Aout, nE);
        }
    };
    auto run_bn = [&](float* F, int layer) {
        k_fill<<<2, 256, 0, stream>>>(sSum, 0.f, (size_t)(2 * DH));  // sSum+sSq contiguous
        k_bn_stats<<<(nN + 511) / 512, 256, 0, stream>>>(F, sSum, sSq, nN, 512);
        size_t tot = (size_t)nN * DH;
        k_bn_apply_relu<<<(unsigned)((tot + 255) / 256), 256, 0, stream>>>(
            F, sSum, sSq, gamma + (size_t)layer * DH, beta + (size_t)layer * DH, nN);
    };

    // layer 0: x(128) -> B2 -> B1, BN0
    run_conv(x, DIN, W0, DH, B2, B1);
    run_bn(B1, 0);
    // hidden layers 1..3 (B1 is dead once its GEMM into B2 completes)
    for (int i = 0; i < 3; ++i) {
        run_conv(B1, DH, Wh + (size_t)i * DH * DH, DH, B2, B1);
        run_bn(B1, i + 1);
    }
    // final 256 -> 40 conv, then bias + log-softmax into d_out
    run_conv(B1, DH, Wl, DOUT, B2, B1);
    k_bias_logsoftmax<<<(nN + 7) / 8, 256, 0, stream>>>(B1, bl, (float*)d_out, nN);
}